// QuantumCBOW_18786186952932
// MI455X (gfx1250) — compile-verified
//
#include <hip/hip_runtime.h>
#include <math.h>

typedef float v8f __attribute__((ext_vector_type(8)));
typedef float v2f __attribute__((ext_vector_type(2)));

#define QDIM 8
#define QTRIL 36
#define QEPS 1e-6f
#define QCADD 2e-6f   // EPS + deterministic corr (rho is PSD by construction)

// fast HW approximations (v_rcp_f32 / v_rsq_f32 / v_sqrt_f32): fine for Jacobi
__device__ __forceinline__ float frcp(float x)  { return __builtin_amdgcn_rcpf(x); }
__device__ __forceinline__ float frsq(float x)  { return __builtin_amdgcn_rsqf(x); }
__device__ __forceinline__ float fsqrt(float x) { return __builtin_amdgcn_sqrtf(x); }

// ---------- DPP8 cross-lane helpers (all exchanges stay in 8-lane groups) ----------
constexpr unsigned dpp8_bcast(int p) {
    unsigned s = 0;
    for (int i = 0; i < 8; ++i) s |= (unsigned)p << (3 * i);
    return s;
}
constexpr unsigned dpp8_swap(int p, int q) {
    unsigned s = 0;
    for (int i = 0; i < 8; ++i) {
        int sel = (i == p) ? q : ((i == q) ? p : i);
        s |= (unsigned)sel << (3 * i);
    }
    return s;
}
constexpr unsigned dpp8_xor(int m) {
    unsigned s = 0;
    for (int i = 0; i < 8; ++i) s |= (unsigned)(i ^ m) << (3 * i);
    return s;
}

template <unsigned SEL>
__device__ __forceinline__ float dpp8f(float x) {
    return __int_as_float(__builtin_amdgcn_mov_dpp8(__float_as_int(x), SEL));
}

// lane n's own diagonal element a[n] (compile-time select chain, no shuffle)
__device__ __forceinline__ float diag_of(const float a[8], int n) {
    float d = 0.0f;
#pragma unroll
    for (int i = 0; i < 8; ++i) d += (n == i) ? a[i] : 0.0f;
    return d;
}

// L[r][c] from packed tril params, diag clamped >= 1e-4
__device__ __forceinline__ float l_elem(const float* __restrict__ p, int r, int c) {
    float v = (c <= r) ? p[(r * (r + 1)) / 2 + c] : 0.0f;
    if (c == r) v = fmaxf(v, 1e-4f);
    return (c <= r) ? v : 0.0f;
}

// D = [[A0],[A1]] (16x8) @ [B0|B1] (8x16) via two V_WMMA_F32_16X16X4_F32.
// A0/A1/B0/B1 are 8x8 column-major in LDS. transB: use B^T.
// Result: lanes 0-7 hold columns of A0*B0 (regs = rows), lanes 24-31 hold A1*B1.
__device__ __forceinline__ v8f wmma_pair8(const float* A0, const float* A1,
                                          const float* B0, const float* B1,
                                          bool transB, int lane) {
    int half = lane >> 4;
    int rn   = lane & 15;
    const float* Am = (rn < 8) ? A0 : A1;
    const float* Bm = (rn < 8) ? B0 : B1;
    int rr = rn & 7;
    v8f c = {0.f, 0.f, 0.f, 0.f, 0.f, 0.f, 0.f, 0.f};
#pragma unroll
    for (int k0 = 0; k0 < 8; k0 += 4) {
        int k = k0 + 2 * half;
        v2f a, b;
        a.x = Am[k * 8 + rr];
        a.y = Am[(k + 1) * 8 + rr];
        if (transB) {
            b.x = Bm[k * 8 + rr];
            b.y = Bm[(k + 1) * 8 + rr];
        } else {
            b.x = Bm[rr * 8 + k];
            b.y = Bm[rr * 8 + (k + 1)];
        }
        c = __builtin_amdgcn_wmma_f32_16x16x4_f32(false, a, false, b,
                                                  (short)0, c, false, false);
    }
    return c;
}

// rho'' = LL^T + (EPS+1e-6)I ; rho = rho'' / (tr + EPS). Column-per-lane layout.
__device__ __forceinline__ void normalize_rho(v8f& c, int n) {
    float dsum = 0.f;
#pragma unroll
    for (int i = 0; i < 8; ++i) {
        float e = c[i] + ((n == i) ? QCADD : 0.0f);
        c[i] = e;
        dsum += (n == i) ? e : 0.0f;
    }
    dsum += dpp8f<dpp8_xor(1)>(dsum);
    dsum += dpp8f<dpp8_xor(2)>(dsum);
    dsum += dpp8f<dpp8_xor(4)>(dsum);
    float inv = frcp(dsum + QEPS);
#pragma unroll
    for (int i = 0; i < 8; ++i) c[i] *= inv;
}

// One Jacobi rotation with compile-time (P,Q): pure VALU via DPP8, no IEEE divides.
template <bool WITHV, int P, int Q>
__device__ __forceinline__ void rot8(float a[8], float v[8], int n) {
    float app = dpp8f<dpp8_bcast(P)>(a[P]);
    float aqq = dpp8f<dpp8_bcast(Q)>(a[Q]);
    float apq = dpp8f<dpp8_bcast(Q)>(a[P]);
    float tau = (aqq - app) * frcp(apq + apq);
    float t   = copysignf(1.0f, tau) * frcp(fabsf(tau) + fsqrt(fmaf(tau, tau, 1.0f)));
    t = (fabsf(apq) < 1e-30f) ? 0.0f : t;        // apq==0 -> identity rotation
    float cc = frsq(fmaf(t, t, 1.0f));
    float ss = t * cc;
    bool isP = (n == P), isQ = (n == Q);
    // per-lane blend coefficients: lanes P,Q mix columns; others pass through
    float alpha = (isP || isQ) ? cc : 1.0f;
    float beta  = isP ? -ss : (isQ ? ss : 0.0f);
    // column mix (A <- A*R): lanes P,Q exchange columns via dpp8 swap
#pragma unroll
    for (int i = 0; i < 8; ++i) {
        float oth = dpp8f<dpp8_swap(P, Q)>(a[i]);
        a[i] = fmaf(beta, oth, alpha * a[i]);
    }
    // row mix (A <- R^T * A)
    float tp = a[P], tq = a[Q];
    a[P] = fmaf(cc, tp, -ss * tq);
    a[Q] = fmaf(ss, tp, cc * tq);
    if (WITHV) {
#pragma unroll
        for (int i = 0; i < 8; ++i) {
            float oth = dpp8f<dpp8_swap(P, Q)>(v[i]);
            v[i] = fmaf(beta, oth, alpha * v[i]);
        }
    }
}

// Cyclic Jacobi on symmetric 8x8; lane n (groups at lanes 0-7 / 24-31) holds column n.
template <bool WITHV>
__device__ void jacobi8(float a[8], float v[8], int n, int sweeps) {
    if (WITHV) {
#pragma unroll
        for (int i = 0; i < 8; ++i) v[i] = (i == n) ? 1.0f : 0.0f;
    }
    for (int sw = 0; sw < sweeps; ++sw) {
#define R_(P, Q) rot8<WITHV, P, Q>(a, v, n)
        R_(0,1); R_(0,2); R_(0,3); R_(0,4); R_(0,5); R_(0,6); R_(0,7);
        R_(1,2); R_(1,3); R_(1,4); R_(1,5); R_(1,6); R_(1,7);
        R_(2,3); R_(2,4); R_(2,5); R_(2,6); R_(2,7);
        R_(3,4); R_(3,5); R_(3,6); R_(3,7);
        R_(4,5); R_(4,6); R_(4,7);
        R_(5,6); R_(5,7);
        R_(6,7);
#undef R_
    }
}

__global__ __launch_bounds__(32) void qcbow_fidelity_kernel(
    const int* __restrict__ contexts, const int* __restrict__ targets,
    const float* __restrict__ emb, float* __restrict__ out, int B, int S) {
    __shared__ float sp0[QTRIL], sp1[QTRIL];
    __shared__ float smA0[64], smA1[64], smB0[64], smB1[64];

    int lane = threadIdx.x;            // 0..31, one wave per block
    int b0 = blockIdx.x * 2;
    int b1 = b0 + 1;
    bool has1 = (b1 < B);
    if (!has1) b1 = b0;

    int n   = lane & 7;                // local column index within 8-lane group
    int grp = lane >> 4;               // 0 -> batch b0 half, 1 -> batch b1 half

    float acc[8];
#pragma unroll
    for (int i = 0; i < 8; ++i) acc[i] = 0.0f;
    float denom = 0.0f;
    float tgt[8];

    // ---- Phase A: 10 context tokens + 1 target token, 2 batches at once ----
    for (int s = 0; s <= S; ++s) {
        int i0, i1;
        if (s < S) {
            i0 = contexts[(size_t)b0 * S + s];
            i1 = contexts[(size_t)b1 * S + s];
        } else {
            i0 = targets[b0];
            i1 = targets[b1];
        }
        // prefetch next token's embedding rows to overlap the gather latency
        if (s < S && lane == 0) {
            int j0 = (s + 1 < S) ? contexts[(size_t)b0 * S + s + 1] : targets[b0];
            int j1 = (s + 1 < S) ? contexts[(size_t)b1 * S + s + 1] : targets[b1];
            __builtin_prefetch(&emb[(size_t)j0 * QTRIL], 0, 0);
            __builtin_prefetch(&emb[(size_t)j1 * QTRIL], 0, 0);
        }
        // cooperative gather of 2 x 36 params into LDS
        sp0[lane] = emb[(size_t)i0 * QTRIL + lane];
        sp1[lane] = emb[(size_t)i1 * QTRIL + lane];
        if (lane < QTRIL - 32) {
            sp0[32 + lane] = emb[(size_t)i0 * QTRIL + 32 + lane];
            sp1[32 + lane] = emb[(size_t)i1 * QTRIL + 32 + lane];
        }
        __syncthreads();
        // build L0, L1 column-major in LDS
#pragma unroll
        for (int e = lane; e < 64; e += 32) {
            int cc = e >> 3, rr = e & 7;
            smA0[e] = l_elem(sp0, rr, cc);
            smA1[e] = l_elem(sp1, rr, cc);
        }
        __syncthreads();
        v8f rho = wmma_pair8(smA0, smA1, smA0, smA1, /*transB=*/true, lane); // L L^T
        normalize_rho(rho, n);
        if (s < S) {
            int   mi = grp ? i1 : i0;
            float m  = (mi != 0) ? 1.0f : 0.0f;
#pragma unroll
            for (int i = 0; i < 8; ++i) acc[i] += m * rho[i];
            denom += m;
        } else {
#pragma unroll
            for (int i = 0; i < 8; ++i) tgt[i] = rho[i];
        }
        __syncthreads();
    }

    // context_rho = masked mean (denom==0 -> 0*inf = NaN, matching reference's 0/0)
    float a[8], v[8];
    float rden = frcp(denom);
#pragma unroll
    for (int i = 0; i < 8; ++i) a[i] = acc[i] * rden;

    // ---- Phase B: sqrt_rho via eigendecomposition ----
    jacobi8<true>(a, v, n, 6);
    float wn = diag_of(a, n);                    // lane's own eigenvalue
    float sc = fsqrt(fsqrt(fabsf(wn) + QEPS));   // (|w_n|+EPS)^(1/4)
#pragma unroll
    for (int i = 0; i < 8; ++i) v[i] *= sc;      // V' = V diag((w+EPS)^(1/4))

    bool active = (lane < 8) || (lane >= 24);
    {
        float* dst = (lane < 8) ? smA0 : smA1;
        if (active)
#pragma unroll
            for (int i = 0; i < 8; ++i) dst[n * 8 + i] = v[i];
    }
    __syncthreads();
    v8f R = wmma_pair8(smA0, smA1, smA0, smA1, /*transB=*/true, lane); // V' V'^T
    __syncthreads();
    {
        float* dA = (lane < 8) ? smA0 : smA1;
        float* dB = (lane < 8) ? smB0 : smB1;
        if (active)
#pragma unroll
            for (int i = 0; i < 8; ++i) { dA[n * 8 + i] = R[i]; dB[n * 8 + i] = tgt[i]; }
    }
    __syncthreads();
    v8f T = wmma_pair8(smA0, smA1, smB0, smB1, /*transB=*/false, lane); // T = R*sigma
    __syncthreads();
    {
        float* dB = (lane < 8) ? smB0 : smB1;
        if (active)
#pragma unroll
            for (int i = 0; i < 8; ++i) dB[n * 8 + i] = T[i];
    }
    __syncthreads();
    v8f P = wmma_pair8(smB0, smB1, smA0, smA1, /*transB=*/false, lane); // P = T*R

    float pa[8], dummy[8];
#pragma unroll
    for (int i = 0; i < 8; ++i) pa[i] = P[i];
    jacobi8<false>(pa, dummy, n, 6);

    // f = sum sqrt(|eig| + EPS): per-lane own diagonal, then xor-DPP8 group reduce
    float g = fsqrt(fabsf(diag_of(pa, n)) + QEPS);
    g += dpp8f<dpp8_xor(1)>(g);
    g += dpp8f<dpp8_xor(2)>(g);
    g += dpp8f<dpp8_xor(4)>(g);
    float f = fminf(g, 1.0f);
    f = fmaxf(f, 1e-8f);
    float res = -logf(f);

    if (lane == 0) out[b0] = res;
    if (lane == 24 && has1) out[b1] = res;
}

extern "C" void kernel_launch(void* const* d_in, const int* in_sizes, int n_in,
                              void* d_out, int out_size, void* d_ws, size_t ws_size,
                              hipStream_t stream) {
    const int*   contexts = (const int*)d_in[0];
    const int*   targets  = (const int*)d_in[1];
    const float* emb      = (const float*)d_in[2];
    float*       out      = (float*)d_out;
    int B = in_sizes[1];
    int S = (B > 0) ? in_sizes[0] / B : 0;
    int blocks = (B + 1) / 2;
    qcbow_fidelity_kernel<<<dim3(blocks), dim3(32), 0, stream>>>(
        contexts, targets, emb, out, B, S);
}